// BahdanauAttention_87265145520380
// MI455X (gfx1250) — compile-verified
//
#include <hip/hip_runtime.h>
#include <math.h>
#include <stdint.h>

typedef __attribute__((ext_vector_type(2))) float v2f;
typedef __attribute__((ext_vector_type(8))) float v8f;

#define S_LEN 4096
#define N_B   32
#define H_DIM 512

// Hardware transcendentals (gfx1250 V_TANH_F32 / V_EXP_F32), guarded.
#if __has_builtin(__builtin_amdgcn_tanhf)
  #define FAST_TANH(x) __builtin_amdgcn_tanhf(x)
#elif __has_builtin(__builtin_amdgcn_tanh_f32)
  #define FAST_TANH(x) __builtin_amdgcn_tanh_f32(x)
#else
  #define FAST_TANH(x) tanhf(x)
#endif

#if __has_builtin(__builtin_amdgcn_exp2f)
  #define FAST_EXP(x) __builtin_amdgcn_exp2f((x) * 1.4426950408889634f)
#else
  #define FAST_EXP(x) expf(x)
#endif

// LDS B-panel geometry: 256 U_w rows x 16 k-columns, padded row stride 20 dwords
#define PANEL_W    16
#define PANEL_PADW 20
#define N_PANELS   32          // 32 panels of 16 k-columns per 256-col pass

// d_ws float-offset layout
#define WS_WH     0          // 16384 floats  : wh[N][H]
#define WS_SCORE  16384      // 131072 floats : raw scores [S][N]
#define WS_PART   147456     // 262144 floats : context partials [16][N][H]

// ---------------------------------------------------------------------------
// Kernel 1: wh = hidden @ W_w^T + W_b   ([32,512] -- trivial cost)
// ---------------------------------------------------------------------------
__global__ __launch_bounds__(256) void wh_kernel(const float* __restrict__ hidden,
                                                 const float* __restrict__ W_w,
                                                 const float* __restrict__ W_b,
                                                 float* __restrict__ wh) {
    const int idx = blockIdx.x * 256 + threadIdx.x;   // [0, 16384)
    const int n = idx >> 9;
    const int k = idx & 511;
    const float* hrow = hidden + (size_t)n * H_DIM;
    const float* wrow = W_w + (size_t)k * H_DIM;
    float acc = W_b[k];
#pragma unroll 8
    for (int h = 0; h < H_DIM; ++h) acc += hrow[h] * wrow[h];
    wh[idx] = acc;
}

// ---------------------------------------------------------------------------
// Async VGPR-bypass copy of one U_w panel into LDS (4 x b128 per thread).
// global_load_async_to_lds_b128: VDST = LDS byte address, VADDR = 64-bit
// global address; tracked with ASYNCcnt (CDNA5 ISA).
// ---------------------------------------------------------------------------
__device__ __forceinline__ void prefetch_panel_async(const float* __restrict__ Uw,
                                                     int pass, int panel,
                                                     float* ldsDst, int tid) {
#pragma unroll
    for (int i = 0; i < 4; ++i) {
        const int e  = i * 256 + tid;            // 1024 float4s total
        const int np = e >> 2;                   // U_w row within 256-row panel
        const int c4 = e & 3;                    // float4 within 16-col panel
        const float* g = Uw + (size_t)(pass * 256 + np) * H_DIM + panel * PANEL_W + c4 * 4;
        const uint32_t l = (uint32_t)(uintptr_t)(ldsDst + np * PANEL_PADW + c4 * 4);
        asm volatile("global_load_async_to_lds_b128 %0, %1, off"
                     :: "v"(l), "v"(g) : "memory");
    }
}

// ---------------------------------------------------------------------------
// Kernel 2: fused  score[s,n] = v_w . tanh(wh[n] + enc[s,n]@U_w^T + U_b) + v_b
// One block = 8 waves = 128 rows of (s,n). WMMA f32 16x16x4.
//   A (16x4)  : lane = (M = lane&15), VGPR g & lane-half lg give K = 2*lg + g
//   B (4x16)  : lane = (N = lane&15), same K mapping  -> B[k][n'] = U_w[n'][k]
//   C/D(16x16): VGPR g holds row M = g + 8*lg, column N = lane&15
// U_w panels double-buffered in LDS via async-to-LDS copies.
// ---------------------------------------------------------------------------
__global__ __launch_bounds__(256) void score_kernel(const float* __restrict__ enc,
                                                    const float* __restrict__ U_w,
                                                    const float* __restrict__ U_b,
                                                    const float* __restrict__ v_w,
                                                    const float* __restrict__ v_b,
                                                    const float* __restrict__ wh,
                                                    float* __restrict__ score) {
    __shared__ float ldsB[2][256 * PANEL_PADW];       // 2 x 20 KiB
    const int tid  = threadIdx.x;
    const int wave = tid >> 5;
    const int lane = tid & 31;
    const int m    = lane & 15;                       // M / N' position
    const int lg   = lane >> 4;                       // lane half
    const int rowBase = blockIdx.x * 128 + wave * 16; // first global row of tile

    float spart[8] = {0.f,0.f,0.f,0.f,0.f,0.f,0.f,0.f};
    const v8f vzero = {0.f,0.f,0.f,0.f,0.f,0.f,0.f,0.f};

    for (int pass = 0; pass < 2; ++pass) {            // 2 x 256 output columns
        v8f acc[16];
#pragma unroll
        for (int t = 0; t < 16; ++t) acc[t] = vzero;

        // preamble: stage panel 0 into buffer 0
        prefetch_panel_async(U_w, pass, 0, &ldsB[0][0], tid);

        for (int panel = 0; panel < N_PANELS; ++panel) {
            const int cur = panel & 1;
            if (panel < N_PANELS - 1) {
                // overlap: issue next panel's copy into the other buffer
                prefetch_panel_async(U_w, pass, panel + 1, &ldsB[1 - cur][0], tid);
                // async loads retire in order: <=4 outstanding => current panel done
                asm volatile("s_wait_asynccnt 0x4" ::: "memory");
            } else {
                asm volatile("s_wait_asynccnt 0x0" ::: "memory");
            }
            __syncthreads();                          // all waves' copies visible

            const float* bbuf = &ldsB[cur][0];
#pragma unroll
            for (int ks = 0; ks < 4; ++ks) {          // 4 k-steps of 4
                const int k0 = panel * PANEL_W + ks * 4;
                const v2f a = *(const v2f*)(enc +
                        (size_t)(rowBase + m) * H_DIM + k0 + lg * 2);
#pragma unroll
                for (int t = 0; t < 16; ++t) {        // 16 n'-tiles
                    const v2f b = *(const v2f*)(bbuf + (t * 16 + m) * PANEL_PADW + ks * 4 + lg * 2);
                    acc[t] = __builtin_amdgcn_wmma_f32_16x16x4_f32(
                        /*neg_a=*/false, a, /*neg_b=*/false, b,
                        /*c_mod=*/(short)0, acc[t],
                        /*reuse_a=*/false, /*reuse_b=*/false);
                }
            }
            __syncthreads();                          // buffer consumed by all waves
        }

        // fused epilogue: hardware tanh + dot with v_w (partial per row)
#pragma unroll
        for (int t = 0; t < 16; ++t) {
            const int col = pass * 256 + t * 16 + m;
            const float vw = v_w[col];
            const float ub = U_b[col];
#pragma unroll
            for (int g = 0; g < 8; ++g) {
                const int row = rowBase + g + 8 * lg;
                const int n   = row & (N_B - 1);      // row = s*32 + n
                const float e = acc[t][g] + wh[n * H_DIM + col] + ub;
                spart[g] += FAST_TANH(e) * vw;
            }
        }
    }

    // reduce the 16 columns held across lanes of each half-wave
#pragma unroll
    for (int g = 0; g < 8; ++g) {
        float v = spart[g];
        v += __shfl_xor(v, 1);
        v += __shfl_xor(v, 2);
        v += __shfl_xor(v, 4);
        v += __shfl_xor(v, 8);
        spart[g] = v;
    }
    if (m == 0) {                                     // lanes 0 (rows 0..7) and 16 (rows 8..15)
        const float vb = v_b[0];
#pragma unroll
        for (int g = 0; g < 8; ++g)
            score[rowBase + g + 8 * lg] = spart[g] + vb;
    }
}

// ---------------------------------------------------------------------------
// Kernel 3: softmax over S (axis 0) for each n; writes attn into d_out
// ---------------------------------------------------------------------------
__global__ __launch_bounds__(256) void softmax_kernel(const float* __restrict__ score,
                                                      float* __restrict__ attn) {
    const int n   = blockIdx.x;
    const int tid = threadIdx.x;
    __shared__ float red[256];

    float v[16];
    float mx = -3.402823466e38f;
#pragma unroll
    for (int i = 0; i < 16; ++i) {
        v[i] = score[(size_t)(i * 256 + tid) * N_B + n];
        mx = fmaxf(mx, v[i]);
    }
    red[tid] = mx; __syncthreads();
    for (int off = 128; off > 0; off >>= 1) {
        if (tid < off) red[tid] = fmaxf(red[tid], red[tid + off]);
        __syncthreads();
    }
    mx = red[0]; __syncthreads();

    float s = 0.f;
#pragma unroll
    for (int i = 0; i < 16; ++i) { v[i] = FAST_EXP(v[i] - mx); s += v[i]; }
    red[tid] = s; __syncthreads();
    for (int off = 128; off > 0; off >>= 1) {
        if (tid < off) red[tid] += red[tid + off];
        __syncthreads();
    }
    const float inv = 1.0f / red[0];
#pragma unroll
    for (int i = 0; i < 16; ++i)
        attn[(size_t)(i * 256 + tid) * N_B + n] = v[i] * inv;
}

// ---------------------------------------------------------------------------
// Kernel 4a: context partials over 256-s chunks (deterministic, no atomics)
// grid (hc=4, sc=16, n=32), block 128 : partial[sc][n][h] = sum_s attn*enc
// ---------------------------------------------------------------------------
__global__ __launch_bounds__(128) void ctx_partial_kernel(const float* __restrict__ enc,
                                                          const float* __restrict__ attn,
                                                          float* __restrict__ partial) {
    const int hc = blockIdx.x, sc = blockIdx.y, n = blockIdx.z;
    const int h  = hc * 128 + threadIdx.x;
    float acc = 0.f;
    for (int i = 0; i < 256; ++i) {
        const int r = (sc * 256 + i) * N_B + n;
        acc += attn[r] * enc[(size_t)r * H_DIM + h];
    }
    partial[(size_t)(sc * N_B + n) * H_DIM + h] = acc;
}

// Kernel 4b: final reduce of 16 partials -> context[n][h] in d_out
__global__ __launch_bounds__(256) void ctx_reduce_kernel(const float* __restrict__ partial,
                                                         float* __restrict__ ctx) {
    const int idx = blockIdx.x * 256 + threadIdx.x;   // [0, 16384) = n*512 + h
    float s = 0.f;
#pragma unroll
    for (int sc = 0; sc < 16; ++sc)
        s += partial[(size_t)sc * (N_B * H_DIM) + idx];
    ctx[idx] = s;
}

// ---------------------------------------------------------------------------
extern "C" void kernel_launch(void* const* d_in, const int* in_sizes, int n_in,
                              void* d_out, int out_size, void* d_ws, size_t ws_size,
                              hipStream_t stream) {
    const float* hidden = (const float*)d_in[0];
    const float* enc    = (const float*)d_in[1];
    const float* W_w    = (const float*)d_in[2];
    const float* W_b    = (const float*)d_in[3];
    const float* U_w    = (const float*)d_in[4];
    const float* U_b    = (const float*)d_in[5];
    const float* v_w    = (const float*)d_in[6];
    const float* v_b    = (const float*)d_in[7];

    float* out  = (float*)d_out;
    float* ctx  = out;                   // context [32,512]
    float* attn = out + N_B * H_DIM;     // attention_weights [4096,32,1]

    float* ws    = (float*)d_ws;
    float* wh    = ws + WS_WH;
    float* score = ws + WS_SCORE;
    float* part  = ws + WS_PART;

    wh_kernel<<<64, 256, 0, stream>>>(hidden, W_w, W_b, wh);
    score_kernel<<<1024, 256, 0, stream>>>(enc, U_w, U_b, v_w, v_b, wh, score);
    softmax_kernel<<<32, 256, 0, stream>>>(score, attn);
    ctx_partial_kernel<<<dim3(4, 16, 32), 128, 0, stream>>>(enc, attn, part);
    ctx_reduce_kernel<<<64, 256, 0, stream>>>(part, ctx);
}